// TOGL_86019605004897
// MI455X (gfx1250) — compile-verified
//
#include <hip/hip_runtime.h>

// -------- problem constants (match reference) --------
#define N_NODES 4096
#define N_EDGES 32768
#define N_FEAT  1024
#define HIDDEN  4096
#define N_FILT  4

typedef __attribute__((ext_vector_type(2))) float v2f;
typedef __attribute__((ext_vector_type(8))) float v8f;

// ================= Kernel 1: fused MLP with fp32 WMMA =================
// F = relu(X @ W1 + b1) @ W2 + b2, computed without materializing H.
// WG = 256 threads (8 waves). Each WG owns a 32-row strip of X staged in LDS.
// Each wave: one 16-row subtile x 64 hidden 16-col tiles; K-loop of
// v_wmma_f32_16x16x4_f32; ReLU + rank-1 update against W2 folds GEMM2 in.
#define MT   32
#define GT   256
#define ASTR (N_FEAT + 4)   // padded LDS row stride (floats): 1028 % 64 = 4 -> conflict-free A frags

__global__ __launch_bounds__(GT)
void togl_mlp_wmma(const float* __restrict__ X, const float* __restrict__ W1,
                   const float* __restrict__ b1, const float* __restrict__ W2,
                   const float* __restrict__ b2, float* __restrict__ F) {
  extern __shared__ char smem[];
  float* As = (float*)smem;                                   // MT x ASTR
  float* Fs = (float*)(smem + (size_t)MT * ASTR * sizeof(float)); // 8 waves x 16 rows x 4

  const int tid  = threadIdx.x;
  const int wave = tid >> 5;
  const int lane = tid & 31;
  const int hlf  = lane >> 4;     // 0: lanes 0-15, 1: lanes 16-31
  const int l16  = lane & 15;
  const int m0   = blockIdx.x * MT;

  // stage X strip into LDS (float4, coalesced)
  for (int q = tid; q < MT * (N_FEAT / 4); q += GT) {
    int r  = q / (N_FEAT / 4);
    int c4 = q % (N_FEAT / 4);
    ((float4*)(As + (size_t)r * ASTR))[c4] =
        ((const float4*)(X + (size_t)(m0 + r) * N_FEAT))[c4];
  }
  __syncthreads();

  const int msub   = wave >> 2;                    // which 16-row subtile (0/1)
  const int mloc   = msub * 16;
  const int nt_beg = (wave & 3) * (HIDDEN / 16 / 4); // 64 n-tiles per wave
  const int nt_end = nt_beg + (HIDDEN / 16 / 4);

  float acc[8][4];
#pragma unroll
  for (int r = 0; r < 8; ++r)
#pragma unroll
    for (int f = 0; f < 4; ++f) acc[r][f] = 0.f;

  for (int nt = nt_beg; nt < nt_end; ++nt) {
    const int n0 = nt * 16;
    if (nt + 1 < nt_end)  // gfx1250 global_prefetch_b8 hint for next W1 tile
      __builtin_prefetch(W1 + (size_t)(n0 + 16) + l16, 0, 0);

    v8f c = {0.f, 0.f, 0.f, 0.f, 0.f, 0.f, 0.f, 0.f};
#pragma unroll 4
    for (int kk = 0; kk < N_FEAT / 4; ++kk) {
      // A frag (16x4 f32): lanes 0-15 hold K=0,1; lanes 16-31 hold K=2,3
      const float* ap = As + (size_t)(mloc + l16) * ASTR + 4 * kk + 2 * hlf;
      v2f a; a.x = ap[0]; a.y = ap[1];
      // B frag (4x16 f32): N striped across 16-lane halves, K pairs per half
      const float* bp = W1 + (size_t)(4 * kk + 2 * hlf) * HIDDEN + n0 + l16;
      v2f b; b.x = bp[0]; b.y = bp[HIDDEN];
      c = __builtin_amdgcn_wmma_f32_16x16x4_f32(
          /*neg_a=*/false, a, /*neg_b=*/false, b,
          /*c_mod=*/(short)0, c, /*reuse_a=*/false, /*reuse_b=*/false);
    }
    // fuse: relu(H + b1) then rank-1 accumulate against W2 row of this column
    const float  bias = b1[n0 + l16];
    const float4 w2v  = *(const float4*)(W2 + (size_t)(n0 + l16) * N_FILT);
#pragma unroll
    for (int r = 0; r < 8; ++r) {
      float h = c[r] + bias;
      h = h > 0.f ? h : 0.f;
      acc[r][0] += h * w2v.x;
      acc[r][1] += h * w2v.y;
      acc[r][2] += h * w2v.z;
      acc[r][3] += h * w2v.w;
    }
  }

  // reduce over the 16 lanes of each half (each half = different 8 rows)
#pragma unroll
  for (int r = 0; r < 8; ++r)
#pragma unroll
    for (int f = 0; f < 4; ++f) {
      float v = acc[r][f];
      v += __shfl_xor(v, 1);
      v += __shfl_xor(v, 2);
      v += __shfl_xor(v, 4);
      v += __shfl_xor(v, 8);
      acc[r][f] = v;
    }
  if (l16 == 0) {
#pragma unroll
    for (int r = 0; r < 8; ++r)
#pragma unroll
      for (int f = 0; f < 4; ++f)
        Fs[wave * 64 + (hlf * 8 + r) * 4 + f] = acc[r][f];
  }
  __syncthreads();

  // deterministic cross-wave reduce (fixed order, no float atomics) + b2
  if (tid < MT * N_FILT) {
    const int row = tid >> 2;         // 0..31
    const int f   = tid & 3;
    const int ms  = row >> 4;
    float s = b2[f];
#pragma unroll
    for (int q = 0; q < 4; ++q)
      s += Fs[(ms * 4 + q) * 64 + (row & 15) * 4 + f];
    F[(size_t)(m0 + row) * N_FILT + f] = s;
  }
}

// ================= Kernel 2: 0-dim persistence diagram =================
// One workgroup (1024 threads) per filtration. All state in LDS (176 KB):
//   rnk[4096] i32 | parent[4096] i32 | death[4096] i32 | ekey[32768] u32
//   (node-sort u64 buffer aliases the ekey region)
#define PT 1024

__global__ __launch_bounds__(PT)
void togl_pd0(const float* __restrict__ F, const int* __restrict__ eu,
              const int* __restrict__ ev, float* __restrict__ diag) {
  extern __shared__ char smem[];
  int*      rnk    = (int*)smem;                  // 16 KB
  int*      parent = (int*)(smem + 16384);        // 16 KB
  int*      death  = (int*)(smem + 32768);        // 16 KB
  unsigned* ekey   = (unsigned*)(smem + 49152);   // 128 KB
  unsigned long long* nkey = (unsigned long long*)(smem + 49152); // alias, 32 KB

  const int filt = blockIdx.x;
  const int tid  = threadIdx.x;

  // --- node keys: stable ascending by (value, index), float -> sortable uint
  for (int i = tid; i < N_NODES; i += PT) {
    unsigned b = __float_as_uint(F[(size_t)i * N_FILT + filt]);
    b ^= (b & 0x80000000u) ? 0xFFFFFFFFu : 0x80000000u;
    nkey[i] = ((unsigned long long)b << 32) | (unsigned)i;
  }
  __syncthreads();

  // --- bitonic sort 4096 u64 keys in LDS
  for (int k = 2; k <= N_NODES; k <<= 1) {
    for (int j = k >> 1; j > 0; j >>= 1) {
      for (int i = tid; i < N_NODES; i += PT) {
        int ixj = i ^ j;
        if (ixj > i) {
          unsigned long long a = nkey[i], c = nkey[ixj];
          bool up = ((i & k) == 0);
          if ((a > c) == up) { nkey[i] = c; nkey[ixj] = a; }
        }
      }
      __syncthreads();
    }
  }

  // rnk[node] = activation step
  for (int p = tid; p < N_NODES; p += PT)
    rnk[(int)(nkey[p] & 0xFFFFFFFFull)] = p;
  __syncthreads();

  // --- edge keys: stable ascending by (estep, edge index); overwrites nkey
  for (int e = tid; e < N_EDGES; e += PT) {
    int ru = rnk[eu[e]], rv = rnk[ev[e]];
    int s  = ru > rv ? ru : rv;
    ekey[e] = ((unsigned)s << 15) | (unsigned)e;   // 12b step | 15b idx
  }
  __syncthreads();

  // --- bitonic sort 32768 u32 keys in LDS
  for (int k = 2; k <= N_EDGES; k <<= 1) {
    for (int j = k >> 1; j > 0; j >>= 1) {
      for (int i = tid; i < N_EDGES; i += PT) {
        int ixj = i ^ j;
        if (ixj > i) {
          unsigned a = ekey[i], c = ekey[ixj];
          bool up = ((i & k) == 0);
          if ((a > c) == up) { ekey[i] = c; ekey[ixj] = a; }
        }
      }
      __syncthreads();
    }
  }

  for (int i = tid; i < N_NODES; i += PT) { parent[i] = i; death[i] = 0; }
  __syncthreads();

  // --- sequential union-find scan (elder rule), path halving; thread 0
  if (tid == 0) {
    for (int t = 0; t < N_EDGES; ++t) {
      const unsigned key = ekey[t];
      const int e = (int)(key & 0x7FFFu);
      const int s = (int)(key >> 15);
      int a = eu[e], b = ev[e];
      while (parent[a] != a) { int g = parent[parent[a]]; parent[a] = g; a = g; }
      while (parent[b] != b) { int g = parent[parent[b]]; parent[b] = g; b = g; }
      if (a != b) {
        const bool sw  = rnk[b] < rnk[a];   // elder rule: smaller rank survives
        const int  eld = sw ? b : a;
        const int  yng = sw ? a : b;
        parent[yng] = eld;
        death[rnk[yng]] = s;
      }
    }
  }
  __syncthreads();

  // --- finalize + write (ints stored as exactly-representable floats)
  for (int p = tid; p < N_NODES; p += PT) {
    int d = death[p];
    if (d == 0) d = N_NODES - 1;
    diag[((size_t)filt * N_NODES + p) * 2 + 0] = (float)p;
    diag[((size_t)filt * N_NODES + p) * 2 + 1] = (float)d;
  }
}

// ============================ launcher ============================
extern "C" void kernel_launch(void* const* d_in, const int* in_sizes, int n_in,
                              void* d_out, int out_size, void* d_ws, size_t ws_size,
                              hipStream_t stream) {
  const float* X  = (const float*)d_in[0];
  const int*   EL = (const int*)d_in[1];     // (2, N_EDGES) flat
  const float* W1 = (const float*)d_in[2];
  const float* b1 = (const float*)d_in[3];
  const float* W2 = (const float*)d_in[4];
  const float* b2 = (const float*)d_in[5];

  float* F    = (float*)d_out;                       // (4096, 4)
  float* diag = F + (size_t)N_NODES * N_FILT;        // (4, 4096, 2) as floats

  const size_t smem1 = (size_t)MT * ASTR * sizeof(float) + 8 * 16 * 4 * sizeof(float);
  togl_mlp_wmma<<<N_NODES / MT, GT, smem1, stream>>>(X, W1, b1, W2, b2, F);

  const size_t smem2 = 49152 + (size_t)N_EDGES * sizeof(unsigned); // 176 KB LDS
  togl_pd0<<<N_FILT, PT, smem2, stream>>>(F, EL, EL + N_EDGES, diag);
}